// PostProcess_86457691668628
// MI455X (gfx1250) — compile-verified
//
#include <hip/hip_runtime.h>
#include <hip/hip_bf16.h>

typedef __attribute__((ext_vector_type(2))) float    v2f;
typedef __attribute__((ext_vector_type(8))) float    v8f;
typedef __attribute__((ext_vector_type(4))) unsigned u32x4;
typedef __attribute__((ext_vector_type(8))) int      i32x8;
typedef __attribute__((ext_vector_type(4))) int      i32x4;

constexpr int QN    = 900;
constexpr int CN    = 91;
constexpr int NSEL  = 300;
constexpr int NFLAT = QN * CN;          // 81900
constexpr int MH    = 128;
constexpr int MW    = 128;
constexpr int OH    = 512;
constexpr int OW    = 512;

// d_out flat layout (scores, labels, boxes, uncertainty, masks)
constexpr size_t OFF_SCORES = 0;
constexpr size_t OFF_LABELS = 600;
constexpr size_t OFF_BOXES  = 1200;
constexpr size_t OFF_UNC    = 3600;
constexpr size_t OFF_MASKS  = 4200;

__device__ __forceinline__ unsigned sortKeyF(float f) {
  unsigned b = __float_as_uint(f);
  return (b & 0x80000000u) ? ~b : (b | 0x80000000u);
}
__device__ __forceinline__ float unsortKeyF(unsigned u) {
  return (u & 0x80000000u) ? __uint_as_float(u ^ 0x80000000u)
                           : __uint_as_float(~u);
}
__device__ __forceinline__ float sigmoidf_(float x) {
  return 1.0f / (1.0f + expf(-x));
}

// bilinear weight of input row/col r for output index i (scale 4, half-pixel
// centers, edge clamp == jax.image.resize bilinear after renormalization)
__device__ __forceinline__ float wrow(int i, int r) {
  float yf = (i + 0.5f) * 0.25f - 0.5f;
  yf = fminf(fmaxf(yf, 0.0f), 127.0f);
  int r0 = (int)yf;                 // yf >= 0, trunc == floor
  float w1 = yf - (float)r0;
  int r1 = r0 + 1; r1 = r1 > 127 ? 127 : r1;
  float w = 0.0f;
  if (r == r0) w += 1.0f - w1;
  if (r == r1) w += w1;
  return w;
}

// ---------------------------------------------------------------------------
// Kernel 1: per-batch top-300 via radix-select + bitonic sort; writes scores,
// labels, boxes, uncertainty; stashes selected query index in d_ws.
// ---------------------------------------------------------------------------
__global__ void __launch_bounds__(1024)
topk_kernel(const float* __restrict__ logits,
            const float* __restrict__ pboxes,
            const float* __restrict__ plogvars,
            const int*   __restrict__ tsizes,
            float* __restrict__ out,
            int*   __restrict__ qsel) {
  const int b   = blockIdx.x;
  const int tid = threadIdx.x;
  const float* lg = logits + (size_t)b * NFLAT;

  __shared__ unsigned hist[256];
  __shared__ unsigned sh_pref;
  __shared__ unsigned sh_remaining;
  __shared__ unsigned cntG, cntE;
  __shared__ unsigned eqIdxArr[256];
  __shared__ unsigned long long keys[512];

  if (tid == 0) { sh_pref = 0u; sh_remaining = NSEL; }
  __syncthreads();

  // 4x 8-bit radix passes from MSB: find threshold key T of the 300th value
  for (int p = 3; p >= 0; --p) {
    if (tid < 256) hist[tid] = 0u;
    __syncthreads();
    const int shift = 8 * p;
    const unsigned pref = sh_pref;
    for (int i = tid; i < NFLAT; i += blockDim.x) {
      unsigned u = sortKeyF(sigmoidf_(lg[i]));
      bool match = (p == 3) || ((u >> (shift + 8)) == pref);
      if (match) atomicAdd(&hist[(u >> shift) & 255u], 1u);
    }
    __syncthreads();
    if (tid == 0) {
      unsigned cum = 0, rem = sh_remaining;
      for (int bin = 255; bin >= 0; --bin) {
        unsigned h = hist[bin];
        if (cum + h >= rem) {
          sh_pref      = (sh_pref << 8) | (unsigned)bin;
          sh_remaining = rem - cum;
          break;
        }
        cum += h;
      }
    }
    __syncthreads();
  }

  const unsigned Tth      = sh_pref;
  const unsigned needEq   = sh_remaining;       // # of ==T entries to take
  const unsigned nGreater = NSEL - needEq;      // # strictly greater

  if (tid == 0) { cntG = 0u; cntE = 0u; }
  for (int i = tid; i < 512; i += blockDim.x) keys[i] = 0ull;
  __syncthreads();

  // compaction
  for (int i = tid; i < NFLAT; i += blockDim.x) {
    unsigned u = sortKeyF(sigmoidf_(lg[i]));
    if (u > Tth) {
      unsigned pos = atomicAdd(&cntG, 1u);
      if (pos < (unsigned)NSEL)
        keys[pos] = ((unsigned long long)u << 32) | (unsigned)(~(unsigned)i);
    } else if (u == Tth) {
      unsigned pos = atomicAdd(&cntE, 1u);
      if (pos < 256u) eqIdxArr[pos] = (unsigned)i;
    }
  }
  __syncthreads();

  // deterministically take the needEq smallest indices among the ==T ties
  if (tid == 0) {
    unsigned cnt = cntE < 256u ? cntE : 256u;
    unsigned ne  = needEq < cnt ? needEq : cnt;
    for (unsigned s = 0; s < ne; ++s) {
      unsigned best = s;
      for (unsigned t2 = s + 1; t2 < cnt; ++t2)
        if (eqIdxArr[t2] < eqIdxArr[best]) best = t2;
      unsigned tmp = eqIdxArr[s]; eqIdxArr[s] = eqIdxArr[best]; eqIdxArr[best] = tmp;
      keys[nGreater + s] =
          ((unsigned long long)Tth << 32) | (unsigned)(~eqIdxArr[s]);
    }
  }
  __syncthreads();

  // bitonic sort, 512 elems, descending => value desc, index asc on ties
  for (unsigned k = 2; k <= 512u; k <<= 1) {
    for (unsigned j = k >> 1; j > 0; j >>= 1) {
      if (tid < 512) {
        unsigned t = tid, ixj = t ^ j;
        if (ixj > t) {
          bool dirDesc = ((t & k) == 0u);
          unsigned long long a = keys[t], c2 = keys[ixj];
          if (dirDesc ? (a < c2) : (a > c2)) { keys[t] = c2; keys[ixj] = a; }
        }
      }
      __syncthreads();
    }
  }

  // small outputs
  if (tid < NSEL) {
    unsigned long long kk = keys[tid];
    unsigned u   = (unsigned)(kk >> 32);
    unsigned idx = ~(unsigned)(kk & 0xffffffffull);
    int q   = (int)(idx / (unsigned)CN);
    int cls = (int)(idx % (unsigned)CN);

    out[OFF_SCORES + b * NSEL + tid] = unsortKeyF(u);
    out[OFF_LABELS + b * NSEL + tid] = (float)cls;

    const float* bx = pboxes + ((size_t)b * QN + q) * 4;
    float cx = bx[0], cy = bx[1], w = bx[2], h = bx[3];
    float img_h = (float)tsizes[b * 2 + 0];
    float img_w = (float)tsizes[b * 2 + 1];
    float* bo = out + OFF_BOXES + ((size_t)b * NSEL + tid) * 4;
    bo[0] = (cx - 0.5f * w) * img_w;
    bo[1] = (cy - 0.5f * h) * img_h;
    bo[2] = (cx + 0.5f * w) * img_w;
    bo[3] = (cy + 0.5f * h) * img_h;

    const float* lv = plogvars + ((size_t)b * QN + q) * 4;
    out[OFF_UNC + b * NSEL + tid] = 0.25f * (lv[0] + lv[1] + lv[2] + lv[3]);

    qsel[b * NSEL + tid] = q;
  }
}

// ---------------------------------------------------------------------------
// Kernel 2: per selected mask, bilinear 128x128 -> 512x512 + threshold.
//  - TDM (tensor_load_to_lds) DMAs the whole 128x128 f32 tile into LDS once
//  - horizontal pass LDS->LDS (8 input rows x 512 cols slab)
//  - vertical pass as V_WMMA_F32_16X16X4_F32 (2 K=4 fragments per 16x16 tile)
// ---------------------------------------------------------------------------
__global__ void __launch_bounds__(256)
mask_kernel(const float* __restrict__ pmasks,
            const int*   __restrict__ qsel,
            float* __restrict__ out) {
  const int blk = blockIdx.x;           // b*300 + j
  const int b   = blk / NSEL;
  const int tid = threadIdx.x;
  const int q   = qsel[blk];

  float* omask = out + OFF_MASKS + (size_t)blk * (OH * OW);

  __shared__ float Ms[MH * MW];         // 64 KB: source mask tile
  __shared__ float Tw[8 * OW];          // 16 KB: horizontal-pass slab

  // ---- TDM: one DMA of the whole mask tile, issued by wave 0 only ----
  if (tid < 32) {
    const float* Mg = pmasks + (size_t)(b * QN + q) * (MH * MW);
    unsigned long long ga = (unsigned long long)(uintptr_t)Mg;
    unsigned ldsa = (unsigned)(uintptr_t)(void*)Ms;   // low 32 bits = LDS offset

    u32x4 g0;                              // D# group 0 (ISA 8.3)
    g0.x = 1u;                             // count=1, load, no gather
    g0.y = ldsa;                           // lds_addr
    g0.z = (unsigned)ga;                   // global_addr[31:0]
    g0.w = (unsigned)((ga >> 32) & 0x01FFFFFFull) | (2u << 30); // [56:32]|type=2

    i32x8 g1;                              // D# group 1 (ISA 8.4)
    g1[0] = (2 << 16);                     // data_size=4B; no mask/pad/iterate
    g1[1] = (int)((MW & 0xFFFF) << 16);    // tensor_dim0[15:0]
    g1[2] = (int)((MH & 0xFFFF) << 16);    // tensor_dim0[31:16]=0 | dim1[15:0]
    g1[3] = (int)(MW << 16);               // dim1[31:16]=0 | tile_dim0=128
    g1[4] = MH;                            // tile_dim1=128, tile_dim2=0
    g1[5] = MW;                            // tensor_dim0_stride[31:0]=128
    g1[6] = (int)(((MH * MW) & 0xFFFF) << 16); // dim0_str[47:32]=0|dim1_str[15:0]
    g1[7] = (MH * MW) >> 16;               // tensor_dim1_stride[47:16]

    i32x4 g2 = {0, 0, 0, 0};               // 2-D tile: groups 2/3 unused
    i32x4 g3 = {0, 0, 0, 0};
    i32x8 g4 = {0, 0, 0, 0, 0, 0, 0, 0};   // extra group (6-arg builtin form)
    __builtin_amdgcn_tensor_load_to_lds(g0, g1, g2, g3, g4, 0);
    __builtin_amdgcn_s_wait_tensorcnt(0);
  }
  __syncthreads();                         // publish Ms to all waves

  const int lane = tid & 31;
  const int wave = tid >> 5;            // 8 wave32s
  const int half = lane >> 4;           // 0: lanes 0-15, 1: lanes 16-31
  const int mn   = lane & 15;           // M (A) / N (B,D) index
  const int kb   = half * 2;            // A/B fragment K pair base (ISA 7.12.2)

  for (int tr = 0; tr < 32; ++tr) {
    int rbase = 4 * tr - 1;
    rbase = rbase < 0 ? 0 : (rbase > 120 ? 120 : rbase);

    __syncthreads();                    // Tw reuse guard
    // ---- horizontal resample rows rbase..rbase+7 into Tw (LDS -> LDS) ----
    for (int t = tid; t < 8 * OW; t += 256) {
      int k  = t >> 9;
      int co = t & 511;
      float xf = (co + 0.5f) * 0.25f - 0.5f;
      xf = fminf(fmaxf(xf, 0.0f), 127.0f);
      int c0 = (int)xf;
      float w1 = xf - (float)c0;
      int c1 = c0 + 1; c1 = c1 > 127 ? 127 : c1;
      const float* Mr = Ms + (rbase + k) * MW;
      Tw[t] = (1.0f - w1) * Mr[c0] + w1 * Mr[c1];
    }
    __syncthreads();

    // ---- vertical pass via WMMA ----
    const int I0 = tr * 16;
    v2f a0, a1;
    {
      int i = I0 + mn;
      a0.x = wrow(i, rbase + kb);
      a0.y = wrow(i, rbase + kb + 1);
      a1.x = wrow(i, rbase + 4 + kb);
      a1.y = wrow(i, rbase + 4 + kb + 1);
    }
    for (int it = 0; it < 4; ++it) {
      int C0  = (wave + 8 * it) * 16;
      int col = C0 + mn;
      v2f b0, b1;
      b0.x = Tw[(kb    ) * OW + col];
      b0.y = Tw[(kb + 1) * OW + col];
      b1.x = Tw[(kb + 4) * OW + col];
      b1.y = Tw[(kb + 5) * OW + col];
      v8f c = {};
      c = __builtin_amdgcn_wmma_f32_16x16x4_f32(false, a0, false, b0,
                                                (short)0, c, false, false);
      c = __builtin_amdgcn_wmma_f32_16x16x4_f32(false, a1, false, b1,
                                                (short)0, c, false, false);
      float* obase = omask + (size_t)(I0 + half * 8) * OW + col;
#pragma unroll
      for (int v = 0; v < 8; ++v)
        obase[(size_t)v * OW] = c[v] > 0.0f ? 1.0f : 0.0f;
    }
  }
}

extern "C" void kernel_launch(void* const* d_in, const int* in_sizes, int n_in,
                              void* d_out, int out_size, void* d_ws,
                              size_t ws_size, hipStream_t stream) {
  const float* logits = (const float*)d_in[0];   // (2,900,91)
  const float* pboxes = (const float*)d_in[1];   // (2,900,4)
  const float* pmasks = (const float*)d_in[2];   // (2,900,128,128)
  const float* plv    = (const float*)d_in[3];   // (2,900,4)
  const int*   tsz    = (const int*)d_in[4];     // (2,2)
  float* out = (float*)d_out;
  int*   qs  = (int*)d_ws;                       // 600 ints of scratch

  topk_kernel<<<2, 1024, 0, stream>>>(logits, pboxes, plv, tsz, out, qs);
  mask_kernel<<<2 * NSEL, 256, 0, stream>>>(pmasks, qs, out);

  (void)in_sizes; (void)n_in; (void)out_size; (void)ws_size;
}